// SSM_78683800862944
// MI455X (gfx1250) — compile-verified
//
#include <hip/hip_runtime.h>

typedef __attribute__((ext_vector_type(2))) float v2f;
typedef __attribute__((ext_vector_type(8))) float v8f;

#define L_SEQ   2048
#define BATCH   2
#define H_DIM   768
#define N_ST    16
#define R_DIM   48
#define M_ROWS  (L_SEQ * BATCH)        // 4096
#define PW      (R_DIM + 2 * N_ST)     // 80
#define NCHUNK  32
#define LC      (L_SEQ / NCHUNK)       // 64
#define NCH     (BATCH * H_DIM * N_ST) // 24576
#define DT_MIN_F 0.001f
#define DT_MAX_F 0.1f

// ---- async global->LDS copy (CDNA5 GLOBAL_LOAD_ASYNC_TO_LDS_B128) ----------
#if defined(__has_builtin)
#  if __has_builtin(__builtin_amdgcn_global_load_async_to_lds_b128)
#    define HAVE_ASYNC_LDS 1
#  endif
#endif
#ifndef HAVE_ASYNC_LDS
#  define HAVE_ASYNC_LDS 0
#endif

typedef int v4i_async __attribute__((vector_size(16)));
typedef __attribute__((address_space(1))) v4i_async* g_v4i_p;  // global int4*
typedef __attribute__((address_space(3))) v4i_async* l_v4i_p;  // LDS int4*

__device__ __forceinline__ void async_cp16(float* lds_dst, const float* gsrc) {
#if HAVE_ASYNC_LDS
  __builtin_amdgcn_global_load_async_to_lds_b128(
      (g_v4i_p)(const_cast<float*>(gsrc)),
      (l_v4i_p)(lds_dst),
      0, 0);
#else
  *reinterpret_cast<float4*>(lds_dst) = *reinterpret_cast<const float4*>(gsrc);
#endif
}

__device__ __forceinline__ void async_join() {
#if HAVE_ASYNC_LDS
#  if __has_builtin(__builtin_amdgcn_s_wait_asynccnt)
  __builtin_amdgcn_s_wait_asynccnt(0);
#  else
  asm volatile("s_wait_asynccnt 0x0" ::: "memory");
#  endif
#endif
}

// ---------------------------------------------------------------------------
// Kernel 1: proj[M,80] = u[M,768] @ W_ssm[768,80]  (fp32 WMMA 16x16x4)
// Block: 160 threads = 5 waves; wave w computes N-tile w (N=80 = 5 tiles).
// ---------------------------------------------------------------------------
__global__ __launch_bounds__(160)
void ssm_proj_gemm(const float* __restrict__ u,
                   const float* __restrict__ Wssm,
                   float* __restrict__ proj) {
  __shared__ __align__(16) float uS[16 * 68];   // 16 x 64, pitch 68 (bank pad)
  __shared__ __align__(16) float wS[64 * 80];   // 64 x 80

  const int tid   = threadIdx.x;
  const int wv    = tid >> 5;      // wave id -> N tile 0..4
  const int lane  = tid & 31;
  const int laneM = lane & 15;
  const int khalf = lane >> 4;     // 0: K pair {0,1}; 1: K pair {2,3}
  const int mBase = blockIdx.x * 16;

  v8f acc = {};
  for (int k0 = 0; k0 < H_DIM; k0 += 64) {
    for (int i = tid; i < 256; i += 160) {              // 16x64 in 16B chunks
      int r = i >> 4, k = (i & 15) * 4;
      async_cp16(&uS[r * 68 + k], &u[(mBase + r) * H_DIM + k0 + k]);
    }
    for (int i = tid; i < 1280; i += 160) {             // 64x80 in 16B chunks
      int r = i / 20, c = (i - r * 20) * 4;
      async_cp16(&wS[r * 80 + c], &Wssm[(k0 + r) * PW + c]);
    }
    async_join();
    __syncthreads();
    #pragma unroll
    for (int kk = 0; kk < 64; kk += 4) {
      v2f a, b;
      a[0] = uS[laneM * 68 + kk + 2 * khalf + 0];
      a[1] = uS[laneM * 68 + kk + 2 * khalf + 1];
      b[0] = wS[(kk + 2 * khalf + 0) * 80 + wv * 16 + laneM];
      b[1] = wS[(kk + 2 * khalf + 1) * 80 + wv * 16 + laneM];
      acc = __builtin_amdgcn_wmma_f32_16x16x4_f32(false, a, false, b,
                                                  (short)0, acc, false, false);
    }
    __syncthreads();
  }
  // C/D layout: VGPR r -> row (r + 8*khalf), col = lane&15
  const int col = wv * 16 + laneM;
  #pragma unroll
  for (int r = 0; r < 8; ++r) {
    int row = mBase + r + 8 * khalf;
    proj[row * PW + col] = acc[r];
  }
}

// ---------------------------------------------------------------------------
// Kernel 2: delta[M,768] = clip(softplus(proj[:,32:80] @ W_dt[48,768] + b_dt))
// Block: 256 threads = 8 waves; block covers 128 output cols.
// ---------------------------------------------------------------------------
__global__ __launch_bounds__(256)
void ssm_delta_gemm(const float* __restrict__ proj,
                    const float* __restrict__ Wdt,
                    const float* __restrict__ bdt,
                    float* __restrict__ delta) {
  __shared__ __align__(16) float dtS[16 * 52];    // 16 x 48, pitch 52
  __shared__ __align__(16) float wS[48 * 132];    // 48 x 128, pitch 132

  const int tid   = threadIdx.x;
  const int wv    = tid >> 5;       // 0..7
  const int lane  = tid & 31;
  const int laneM = lane & 15;
  const int khalf = lane >> 4;
  const int mBase = blockIdx.x * 16;
  const int colBlock = blockIdx.y * 128;

  for (int i = tid; i < 192; i += 256) {                // 16x48 in 16B chunks
    int r = i / 12, k = (i - r * 12) * 4;
    async_cp16(&dtS[r * 52 + k], &proj[(mBase + r) * PW + 2 * N_ST + k]);
  }
  for (int i = tid; i < 1536; i += 256) {               // 48x128 in 16B chunks
    int r = i >> 5, c = (i & 31) * 4;
    async_cp16(&wS[r * 132 + c], &Wdt[r * H_DIM + colBlock + c]);
  }
  async_join();
  __syncthreads();

  v8f acc = {};
  #pragma unroll
  for (int kk = 0; kk < 48; kk += 4) {
    v2f a, b;
    a[0] = dtS[laneM * 52 + kk + 2 * khalf + 0];
    a[1] = dtS[laneM * 52 + kk + 2 * khalf + 1];
    b[0] = wS[(kk + 2 * khalf + 0) * 132 + wv * 16 + laneM];
    b[1] = wS[(kk + 2 * khalf + 1) * 132 + wv * 16 + laneM];
    acc = __builtin_amdgcn_wmma_f32_16x16x4_f32(false, a, false, b,
                                                (short)0, acc, false, false);
  }
  const int col  = colBlock + wv * 16 + laneM;
  const float bias = bdt[col];
  #pragma unroll
  for (int r = 0; r < 8; ++r) {
    int row = mBase + r + 8 * khalf;
    float x  = acc[r] + bias;
    float sp = __logf(1.0f + __expf(x));   // overflow -> +inf -> clipped below
    sp = fminf(fmaxf(sp, DT_MIN_F), DT_MAX_F);
    delta[row * H_DIM + col] = sp;
  }
}

// ---------------------------------------------------------------------------
// Kernel 3: per-chunk local scan. Block = 256 threads: lane n = tid&15,
// channel ch = tid>>4 (16 h's, all same b). Grid (NCHUNK, B*H/16).
// ---------------------------------------------------------------------------
__global__ __launch_bounds__(256)
void ssm_scan_phase1(const float* __restrict__ delta,
                     const float* __restrict__ u,
                     const float* __restrict__ proj,
                     const float* __restrict__ Alog,
                     float* __restrict__ chA,
                     float* __restrict__ chX) {
  __shared__ __align__(16) float dS[LC * 16], uS[LC * 16], bS[LC * 16];
  const int tid   = threadIdx.x;
  const int n     = tid & 15;
  const int ch    = tid >> 4;
  const int chunk = blockIdx.x;
  const int bh0   = blockIdx.y * 16;
  const int b     = bh0 / H_DIM;
  const int hBase = bh0 - b * H_DIM;
  const int h     = hBase + ch;
  const int l0    = chunk * LC;

  {                                   // 3 x 1024 floats = 3 x (256 x 16B)
    const int i4 = tid * 4;
    const int step = i4 >> 4, j = i4 & 15;
    const int m = (l0 + step) * BATCH + b;
    async_cp16(&dS[i4], &delta[m * H_DIM + hBase + j]);
    async_cp16(&uS[i4], &u[m * H_DIM + hBase + j]);
    async_cp16(&bS[i4], &proj[m * PW + j]);        // Bmat = proj[:, 0:16]
  }
  async_join();
  __syncthreads();

  const float A = -__expf(Alog[h * N_ST + n]);
  float aProd = 1.0f, x = 0.0f;
  #pragma unroll 4
  for (int s = 0; s < LC; ++s) {
    float dv = dS[s * 16 + ch];
    float uv = uS[s * 16 + ch];
    float bv = bS[s * 16 + n];
    float dA = __expf(dv * A);
    x = __fmaf_rn(dA, x, dv * uv * bv);
    aProd *= dA;
  }
  int o = chunk * NCH + bh0 * 16 + tid;     // coalesced
  chA[o] = aProd;
  chX[o] = x;
}

// ---------------------------------------------------------------------------
// Kernel 4: combine chunk summaries -> exclusive carry-in per chunk/channel.
// ---------------------------------------------------------------------------
__global__ __launch_bounds__(256)
void ssm_scan_phase2(const float* __restrict__ chA,
                     const float* __restrict__ chX,
                     float* __restrict__ carry) {
  int cid = blockIdx.x * 256 + threadIdx.x;  // 0..NCH-1
  float c = 0.0f;
  for (int k = 0; k < NCHUNK; ++k) {
    int idx = k * NCH + cid;
    carry[idx] = c;
    c = __fmaf_rn(chA[idx], c, chX[idx]);
  }
}

// ---------------------------------------------------------------------------
// Kernel 5: replay chunk with carry-in, contract over n, write y + u*D.
// ---------------------------------------------------------------------------
__global__ __launch_bounds__(256)
void ssm_scan_phase3(const float* __restrict__ delta,
                     const float* __restrict__ u,
                     const float* __restrict__ proj,
                     const float* __restrict__ Alog,
                     const float* __restrict__ Dv,
                     const float* __restrict__ carry,
                     float* __restrict__ out) {
  __shared__ __align__(16) float dS[LC * 16], uS[LC * 16], bS[LC * 16],
                                 cS[LC * 16], yS[LC * 16];
  const int tid   = threadIdx.x;
  const int n     = tid & 15;
  const int ch    = tid >> 4;
  const int chunk = blockIdx.x;
  const int bh0   = blockIdx.y * 16;
  const int b     = bh0 / H_DIM;
  const int hBase = bh0 - b * H_DIM;
  const int h     = hBase + ch;
  const int l0    = chunk * LC;

  {
    const int i4 = tid * 4;
    const int step = i4 >> 4, j = i4 & 15;
    const int m = (l0 + step) * BATCH + b;
    async_cp16(&dS[i4], &delta[m * H_DIM + hBase + j]);
    async_cp16(&uS[i4], &u[m * H_DIM + hBase + j]);
    async_cp16(&bS[i4], &proj[m * PW + j]);            // Bmat
    async_cp16(&cS[i4], &proj[m * PW + N_ST + j]);     // Cmat = proj[:,16:32]
  }
  async_join();
  __syncthreads();

  const float A  = -__expf(Alog[h * N_ST + n]);
  const float Dh = Dv[h];
  float x = carry[chunk * NCH + bh0 * 16 + tid];
  for (int s = 0; s < LC; ++s) {
    float dv = dS[s * 16 + ch];
    float uv = uS[s * 16 + ch];
    float bv = bS[s * 16 + n];
    float cv = cS[s * 16 + n];
    float dA = __expf(dv * A);
    x = __fmaf_rn(dA, x, dv * uv * bv);
    float yv = cv * x;
    yv += __shfl_xor(yv, 1, 16);
    yv += __shfl_xor(yv, 2, 16);
    yv += __shfl_xor(yv, 4, 16);
    yv += __shfl_xor(yv, 8, 16);
    if (n == 0) yS[s * 16 + ch] = yv + uv * Dh;
  }
  __syncthreads();
  for (int i = tid; i < LC * 16; i += 256) {
    int step = i >> 4, j = i & 15;
    int m = (l0 + step) * BATCH + b;
    out[m * H_DIM + hBase + j] = yS[i];     // coalesced store
  }
}

// ---------------------------------------------------------------------------
extern "C" void kernel_launch(void* const* d_in, const int* in_sizes, int n_in,
                              void* d_out, int out_size, void* d_ws, size_t ws_size,
                              hipStream_t stream) {
  (void)in_sizes; (void)n_in; (void)out_size; (void)ws_size;
  const float* u    = (const float*)d_in[0];
  const float* Alog = (const float*)d_in[1];
  const float* Dv   = (const float*)d_in[2];
  const float* Wssm = (const float*)d_in[3];
  const float* Wdt  = (const float*)d_in[4];
  const float* bdt  = (const float*)d_in[5];
  float* out = (float*)d_out;

  float* ws    = (float*)d_ws;
  float* proj  = ws;                            // 4096*80
  float* delta = proj + (size_t)M_ROWS * PW;    // 4096*768
  float* chA   = delta + (size_t)M_ROWS * H_DIM;
  float* chX   = chA + (size_t)NCHUNK * NCH;
  float* carry = chX + (size_t)NCHUNK * NCH;    // total ~23.3 MB

  ssm_proj_gemm<<<dim3(M_ROWS / 16), dim3(160), 0, stream>>>(u, Wssm, proj);
  ssm_delta_gemm<<<dim3(M_ROWS / 16, H_DIM / 128), dim3(256), 0, stream>>>(proj, Wdt, bdt, delta);
  ssm_scan_phase1<<<dim3(NCHUNK, BATCH * H_DIM / 16), dim3(256), 0, stream>>>(delta, u, proj, Alog, chA, chX);
  ssm_scan_phase2<<<dim3(NCH / 256), dim3(256), 0, stream>>>(chA, chX, carry);
  ssm_scan_phase3<<<dim3(NCHUNK, BATCH * H_DIM / 16), dim3(256), 0, stream>>>(delta, u, proj, Alog, Dv, carry, out);
}